// PanSolver_40492951667278
// MI455X (gfx1250) — compile-verified
//
#include <hip/hip_runtime.h>
#include <math.h>

typedef float v2f __attribute__((ext_vector_type(2)));
typedef float v8f __attribute__((ext_vector_type(8)));

#define NP    64   // collocation points
#define NCF   32   // Chebyshev coeffs
#define NTL   30   // tail coeffs
#define NITER 20

// workspace layout (float offsets)
#define WS_PHI    0                      // [32][80] padded (B-frag bank-friendly)
#define WS_PHIC   (WS_PHI + 32*80)       // [64][40] padded, cols 30..39 zero
#define WS_PTAIL  (WS_PHIC + 64*40)      // [30][2]
#define WS_INV0   (WS_PTAIL + 60)        // [2][2]
#define WS_PHID   (WS_INV0 + 4)          // [2][30]
#define WS_STRUE  (WS_PHID + 60)         // [64]  sin(t_true)
#define WS_PHIOUT (WS_STRUE + 64)        // [32][16]
#define WS_SINT0  (WS_PHIOUT + 512)      // sin(t0)

// Branch-free tanh: t = exp2(-2*log2(e)*|x|) underflows to 0 for large |x|
// (tanh -> 1 exactly); (1-t)/(1+t) via v_rcp; sign restored with v_bfi.
// ~6 VALU + v_exp + v_rcp, no EXEC divergence (vs ~30+ VALU, 2 divergent
// paths for libm tanhf).
__device__ __forceinline__ float fast_tanhf(float x) {
  float ax = __builtin_fabsf(x);
  float t  = __builtin_amdgcn_exp2f(-2.8853900817779268f * ax);
  float r  = (1.0f - t) * __builtin_amdgcn_rcpf(1.0f + t);
  return __builtin_copysignf(r, x);
}

// ---------------------------------------------------------------------------
// Setup: build all t-dependent constant matrices (tiny, runs once per launch)
// ---------------------------------------------------------------------------
__global__ __launch_bounds__(64) void pan_setup(const float* __restrict__ tspan,
                                                int nt, float* __restrict__ ws) {
  __shared__ float sPhi[NCF * NP], sDPhi[NCF * NP];
  __shared__ float sPhiB[NTL * NP], sPhiCt[NP * NTL];
  __shared__ float sMM[NTL * NTL], sQQ[NTL * NTL];
  __shared__ float sDt[NP], sTc[NP], sI0[4];

  const int   tid = threadIdx.x;
  const float t0  = tspan[0];
  const float t1  = tspan[nt - 1];
  const float cc  = 2.0f / (t1 - t0);
  const float PI  = 3.14159265358979323846f;

  { int j = tid; sTc[j] = -cosf(PI * (float)j / (float)NP); }
  __syncthreads();
  {
    int j = tid;
    sDt[j] = ((j < NP - 1) ? sTc[j + 1] : 1.0f) - sTc[j];
    float tc = sTc[j];
    sPhi[0 * NP + j] = 1.0f;  sPhi[1 * NP + j] = tc;
    sDPhi[0 * NP + j] = 0.0f; sDPhi[1 * NP + j] = cc;   // cc * 1 * U0
    float tm2 = 1.0f, tm1 = tc;
    float uprev = 1.0f, ucur = 2.0f * tc;               // U0, U1
    for (int i = 2; i < NCF; i++) {
      float tcu = 2.0f * tc * tm1 - tm2;
      sPhi[i * NP + j] = tcu; tm2 = tm1; tm1 = tcu;
      sDPhi[i * NP + j] = cc * (float)i * ucur;         // cc * i * U_{i-1}
      float un = 2.0f * tc * ucur - uprev; uprev = ucur; ucur = un;
    }
    float tt = t0 + 0.5f * (t1 - t0) * (tc + 1.0f);
    ws[WS_STRUE + j] = sinf(tt);
  }
  __syncthreads();
  if (tid == 0) {
    // inv0 = inv([[Phi00, DPhi00],[Phi10, DPhi10]])
    float a00 = sPhi[0],      a01 = sDPhi[0];
    float a10 = sPhi[1 * NP], a11 = sDPhi[1 * NP];
    float id = 1.0f / (a00 * a11 - a01 * a10);
    sI0[0] =  a11 * id; sI0[1] = -a01 * id;
    sI0[2] = -a10 * id; sI0[3] =  a00 * id;
    ws[WS_INV0 + 0] = sI0[0]; ws[WS_INV0 + 1] = sI0[1];
    ws[WS_INV0 + 2] = sI0[2]; ws[WS_INV0 + 3] = sI0[3];
    ws[WS_SINT0] = sinf(t0);
  }
  if (tid < NTL) {
    ws[WS_PTAIL + 2 * tid]     = sPhi[(tid + 2) * NP];
    ws[WS_PTAIL + 2 * tid + 1] = sDPhi[(tid + 2) * NP];
  }
  __syncthreads();
  {
    int j = tid;  // Phi_b[i][j] = DPhi[i+2][j] - (Phi_tail@inv0@DPhi_head)[i][j]
    for (int i = 0; i < NTL; i++) {
      float pt0 = sPhi[(i + 2) * NP], pt1 = sDPhi[(i + 2) * NP];
      float w0 = pt0 * sI0[0] + pt1 * sI0[2];
      float w1 = pt0 * sI0[1] + pt1 * sI0[3];
      sPhiB[i * NP + j] = sDPhi[(i + 2) * NP + j]
                        - (w0 * sDPhi[0 * NP + j] + w1 * sDPhi[1 * NP + j]);
    }
  }
  __syncthreads();
  for (int idx = tid; idx < NTL * NTL; idx += 64) {   // M = Phi_b @ (Dt * Phi_b^T)
    int i = idx / NTL, k = idx % NTL;
    float s = 0.0f;
    for (int j = 0; j < NP; j++) s += sPhiB[i * NP + j] * sDt[j] * sPhiB[k * NP + j];
    sMM[idx] = s;
  }
  __syncthreads();
  if (tid == 0) {                                      // Q = inv(M), Gauss-Jordan
    for (int r = 0; r < NTL; r++)
      for (int c = 0; c < NTL; c++) sQQ[r * NTL + c] = (r == c) ? 1.0f : 0.0f;
    for (int p = 0; p < NTL; p++) {
      int pr = p; float pm = fabsf(sMM[p * NTL + p]);
      for (int r = p + 1; r < NTL; r++) {
        float v = fabsf(sMM[r * NTL + p]); if (v > pm) { pm = v; pr = r; }
      }
      if (pr != p)
        for (int c = 0; c < NTL; c++) {
          float t = sMM[p * NTL + c]; sMM[p * NTL + c] = sMM[pr * NTL + c]; sMM[pr * NTL + c] = t;
          t = sQQ[p * NTL + c]; sQQ[p * NTL + c] = sQQ[pr * NTL + c]; sQQ[pr * NTL + c] = t;
        }
      float pv = 1.0f / sMM[p * NTL + p];
      for (int c = 0; c < NTL; c++) { sMM[p * NTL + c] *= pv; sQQ[p * NTL + c] *= pv; }
      for (int r = 0; r < NTL; r++) {
        if (r == p) continue;
        float f = sMM[r * NTL + p];
        for (int c = 0; c < NTL; c++) {
          sMM[r * NTL + c] -= f * sMM[p * NTL + c];
          sQQ[r * NTL + c] -= f * sQQ[p * NTL + c];
        }
      }
    }
  }
  __syncthreads();
  for (int idx = tid; idx < NP * NTL; idx += 64) {     // Phi_c = (Dt*Phi_b^T) @ Q
    int j = idx / NTL, i = idx % NTL;
    float s = 0.0f;
    for (int k = 0; k < NTL; k++) s += sPhiB[k * NP + j] * sQQ[k * NTL + i];
    float v = sDt[j] * s;
    sPhiCt[j * NTL + i] = v;
    ws[WS_PHIC + j * 40 + i] = v;
  }
  for (int idx = tid; idx < NP * 10; idx += 64)        // zero pad cols 30..39
    ws[WS_PHIC + (idx / 10) * 40 + 30 + (idx % 10)] = 0.0f;
  __syncthreads();
  for (int idx = tid; idx < 2 * NTL; idx += 64) {      // Phi_d = inv0 @ DPhi_head @ Phi_c
    int m = idx / NTL, i = idx % NTL;
    float e0 = 0.0f, e1 = 0.0f;
    for (int j = 0; j < NP; j++) {
      float pc = sPhiCt[j * NTL + i];
      e0 += sDPhi[0 * NP + j] * pc;
      e1 += sDPhi[1 * NP + j] * pc;
    }
    ws[WS_PHID + m * NTL + i] = sI0[m * 2 + 0] * e0 + sI0[m * 2 + 1] * e1;
  }
  for (int idx = tid; idx < NCF * NP; idx += 64)       // Phi, padded stride 80
    ws[WS_PHI + (idx / NP) * 80 + (idx % NP)] = sPhi[idx];
  if (tid < nt) {                                      // Phi_out columns
    int t = tid;
    float to = -1.0f + 2.0f * (tspan[t] - t0) / (t1 - t0);
    float pm2 = 1.0f, pm1 = to;
    ws[WS_PHIOUT + 0 * nt + t] = 1.0f;
    ws[WS_PHIOUT + 1 * nt + t] = to;
    for (int i = 2; i < NCF; i++) {
      float pc = 2.0f * to * pm1 - pm2;
      ws[WS_PHIOUT + i * nt + t] = pc;
      pm2 = pm1; pm1 = pc;
    }
  }
}

// ---------------------------------------------------------------------------
// Main solver: 4 waves/block, 16 rows per wave, all state LDS-resident.
// GEMMs via V_WMMA_F32_16X16X4_F32 (full f32 precision on the matrix pipe).
// ---------------------------------------------------------------------------
__global__ __launch_bounds__(128) void pan_main(
    const float* __restrict__ y_init, const float* __restrict__ B_init,
    const float* __restrict__ a_vec,  const float* __restrict__ ws,
    float* __restrict__ out, int R, int ncols, int nt)
{
  __shared__ float sPhi[32 * 80];       // B-frags GEMM1 (stride 80: conflict-free)
  __shared__ float sPhiC[64 * 40];      // B-frags GEMM2 (stride 40: conflict-free)
  __shared__ float sStrue[64];
  __shared__ float sPtail[60];
  __shared__ float sInv0[4];
  __shared__ float sPhid[60];
  __shared__ float sPhiout[512];
  __shared__ float sCoeff[4][16 * 34];  // per-wave coeffs (stride 34: A-frag friendly)
  __shared__ float sY[4][16 * 68];      // per-wave grid values (stride 68: A-frag friendly)
  __shared__ float sYfd[4][16 * 30];
  __shared__ float sY0[4][16], sF0[4][16], sA[4][16];

  const int tid  = threadIdx.x;
  const int w    = tid >> 5;
  const int lane = tid & 31;
  const int r0   = blockIdx.x * 64 + w * 16;

  for (int i = tid; i < 32 * 80; i += 128) sPhi[i]  = ws[WS_PHI + i];
  for (int i = tid; i < 64 * 40; i += 128) sPhiC[i] = ws[WS_PHIC + i];
  if (tid < 64) sStrue[tid] = ws[WS_STRUE + tid];
  if (tid < 60) { sPtail[tid] = ws[WS_PTAIL + tid]; sPhid[tid] = ws[WS_PHID + tid]; }
  if (tid < 4)  sInv0[tid] = ws[WS_INV0 + tid];
  for (int i = tid; i < 32 * nt; i += 128) sPhiout[i] = ws[WS_PHIOUT + i];
  const float sinT0 = ws[WS_SINT0];

  // load per-row inputs
  if (lane < 16) {
    int row = r0 + lane;
    float y0 = y_init[row];
    float av = a_vec[row % ncols];
    sY0[w][lane] = y0; sA[w][lane] = av;
    sF0[w][lane] = sinT0 - fast_tanhf(y0 * av);
  }
  for (int idx = lane; idx < 16 * 30; idx += 32) {
    int m = idx / 30, i = idx % 30;
    sCoeff[w][m * 34 + 2 + i] = B_init[(size_t)(r0 + m) * 30 + i];
  }
  __syncthreads();
  for (int idx = lane; idx < 16 * 30; idx += 32) {     // yf_init @ Phi_d
    int m = idx / 30, i = idx % 30;
    sYfd[w][idx] = sY0[w][m] * sPhid[i] + sF0[w][m] * sPhid[30 + i];
  }
  __syncthreads();

  const int mA    = lane & 15;        // A-frag row (M)
  const int koff  = (lane >> 4) * 2;  // A-frag K sub-offset (lanes 16-31: K+2)
  const int mbase = (lane < 16) ? 0 : 8;

  for (int it = 0; it < NITER; it++) {
    // head = (yf - B@Phi_tail) @ inv0   (lanes 0-15, one row each)
    if (lane < 16) {
      float* cf = &sCoeff[w][lane * 34];
      float h0 = sY0[w][lane], h1 = sF0[w][lane];
      #pragma unroll
      for (int i = 0; i < 30; i++) {
        float b = cf[2 + i];
        h0 -= b * sPtail[2 * i]; h1 -= b * sPtail[2 * i + 1];
      }
      cf[0] = h0 * sInv0[0] + h1 * sInv0[2];
      cf[1] = h0 * sInv0[1] + h1 * sInv0[3];
    }
    __syncthreads();

    // GEMM1: y_approx(16x64) = coeff(16x32) @ Phi(32x64), fused tanh
    v2f afr[8];
    #pragma unroll
    for (int kt = 0; kt < 8; kt++) {
      int kb = kt * 4 + koff;
      afr[kt].x = sCoeff[w][mA * 34 + kb];
      afr[kt].y = sCoeff[w][mA * 34 + kb + 1];
    }
    #pragma unroll
    for (int ntile = 0; ntile < 4; ntile++) {
      v8f acc = {0.f, 0.f, 0.f, 0.f, 0.f, 0.f, 0.f, 0.f};
      int nB = ntile * 16 + mA;
      #pragma unroll
      for (int kt = 0; kt < 8; kt++) {
        int kb = kt * 4 + koff;
        v2f b2;
        b2.x = sPhi[kb * 80 + nB];
        b2.y = sPhi[(kb + 1) * 80 + nB];
        acc = __builtin_amdgcn_wmma_f32_16x16x4_f32(
                  false, afr[kt], false, b2, (short)0, acc, false, false);
      }
      float st = sStrue[nB];
      #pragma unroll
      for (int vk = 0; vk < 8; vk++) {       // fap = sin(t_true) - tanh(y*a)
        int m = mbase + vk;
        sY[w][m * 68 + nB] = st - fast_tanhf(acc[vk] * sA[w][m]);
      }
    }
    __syncthreads();

    // GEMM2: B(16x30) = fap(16x64) @ Phi_c(64x30) - yfd
    v2f afr2[16];
    #pragma unroll
    for (int kt = 0; kt < 16; kt++) {
      int kb = kt * 4 + koff;
      afr2[kt].x = sY[w][mA * 68 + kb];
      afr2[kt].y = sY[w][mA * 68 + kb + 1];
    }
    #pragma unroll
    for (int ntile = 0; ntile < 2; ntile++) {
      v8f acc = {0.f, 0.f, 0.f, 0.f, 0.f, 0.f, 0.f, 0.f};
      int nB = ntile * 16 + mA;
      #pragma unroll
      for (int kt = 0; kt < 16; kt++) {
        int kb = kt * 4 + koff;
        v2f b2;
        b2.x = sPhiC[kb * 40 + nB];
        b2.y = sPhiC[(kb + 1) * 40 + nB];
        acc = __builtin_amdgcn_wmma_f32_16x16x4_f32(
                  false, afr2[kt], false, b2, (short)0, acc, false, false);
      }
      if (nB < 30) {
        #pragma unroll
        for (int vk = 0; vk < 8; vk++) {
          int m = mbase + vk;
          sCoeff[w][m * 34 + 2 + nB] = acc[vk] - sYfd[w][m * 30 + nB];
        }
      }
    }
    __syncthreads();
  }

  // final head -> B_full
  if (lane < 16) {
    float* cf = &sCoeff[w][lane * 34];
    float h0 = sY0[w][lane], h1 = sF0[w][lane];
    #pragma unroll
    for (int i = 0; i < 30; i++) {
      float b = cf[2 + i];
      h0 -= b * sPtail[2 * i]; h1 -= b * sPtail[2 * i + 1];
    }
    cf[0] = h0 * sInv0[0] + h1 * sInv0[2];
    cf[1] = h0 * sInv0[1] + h1 * sInv0[3];
  }
  __syncthreads();

  // outputs: approx (nt x R, transposed layout) then B_full (R x 32)
  size_t approxN = (size_t)nt * (size_t)R;
  for (int idx = lane; idx < 16 * 32; idx += 32) {
    int m = idx >> 5, i = idx & 31;
    out[approxN + (size_t)(r0 + m) * 32 + i] = sCoeff[w][m * 34 + i];
  }
  for (int idx = lane; idx < 16 * nt; idx += 32) {
    int m = idx / nt, t = idx % nt;
    float s = 0.0f;
    #pragma unroll
    for (int i = 0; i < 32; i++) s += sCoeff[w][m * 34 + i] * sPhiout[i * nt + t];
    out[(size_t)t * R + (r0 + m)] = s;
  }
}

extern "C" void kernel_launch(void* const* d_in, const int* in_sizes, int n_in,
                              void* d_out, int out_size, void* d_ws, size_t ws_size,
                              hipStream_t stream) {
  const float* y_init = (const float*)d_in[0];
  const float* B_init = (const float*)d_in[1];
  const float* t_span = (const float*)d_in[2];
  const float* a_vec  = (const float*)d_in[3];
  float* out = (float*)d_out;
  float* ws  = (float*)d_ws;

  const int R     = in_sizes[0];   // 256*512 rows
  const int NT    = in_sizes[2];   // 16
  const int NCOLS = in_sizes[3];   // 512

  pan_setup<<<1, 64, 0, stream>>>(t_span, NT, ws);
  pan_main<<<R / 64, 128, 0, stream>>>(y_init, B_init, a_vec, ws, out, R, NCOLS, NT);
}